// NodeModel_2_23630910063283
// MI455X (gfx1250) — compile-verified
//
#include <hip/hip_runtime.h>

typedef __attribute__((ext_vector_type(2))) float v2f;
typedef __attribute__((ext_vector_type(8))) float v8f;

#define WAVES  8     // waves per block
#define ITER   8     // 16-edge tiles per wave
#define NCHUNK 17    // 65 features padded to 17*4
#define PFD    4     // prefetch distance in tiles

// Kernel 1: out[:, :32] = x, out[:, 32:36] = 0 (atomic targets), cnt = 0
__global__ void init_out_kernel(const float* __restrict__ x,
                                float* __restrict__ out,
                                float* __restrict__ cnt, int N) {
    int j = blockIdx.x * blockDim.x + threadIdx.x;
    int total = N * 36;
    if (j < total) {
        int i = j / 36;
        int t = j - i * 36;
        out[j] = (t < 32) ? x[i * 32 + t] : 0.0f;
    }
    if (j < N) cnt[j] = 0.0f;
}

// Kernel 2: each wave processes ITER tiles of 16 edges.
// All A-fragment loads are UNCONDITIONAL (clamped indices) so the compiler
// can cluster them and overlap the whole tile's traffic with the WMMA chain.
__global__ __launch_bounds__(WAVES * 32)
void edge_kernel(const float* __restrict__ x,       // [N, 32]
                 const long long* __restrict__ ei,  // [2, E] int64
                 const float* __restrict__ ea,      // [E, 33]
                 const float* __restrict__ W1,      // [65, 4]
                 const float* __restrict__ b1,      // [4]
                 float* __restrict__ out,           // [N, 36]
                 float* __restrict__ cnt,           // [N]
                 long long E) {
    const int lane = threadIdx.x & 31;
    const int wave = threadIdx.x >> 5;
    const int n    = lane & 15;          // C/D column, B column
    const int hi   = lane >> 4;          // upper half-wave flag
    const int kb   = hi * 2;             // K sub-offset within a 4-chunk
    const int m16  = lane & 15;          // A row / edge-in-tile

    // ---- B fragments: W1 chunks [4 x 16], N-dim zero-padded 4 -> 16 ----
    // (built once per wave, reused for ITER tiles; K=65..67 rows are zero)
    v2f bfrag[NCHUNK];
#pragma unroll
    for (int c = 0; c < NCHUNK; ++c) {
        int k0 = 4 * c + kb;
        float bx = 0.0f, by = 0.0f;
        if (n < 4) {
            if (k0 < 65)     bx = W1[k0 * 4 + n];
            if (k0 + 1 < 65) by = W1[(k0 + 1) * 4 + n];
        }
        bfrag[c].x = bx;
        bfrag[c].y = by;
    }
    const float bias = (n < 4) ? b1[n] : 0.0f;

    const long long tile0 = ((long long)blockIdx.x * WAVES + wave) * ITER;

    for (int i = 0; i < ITER; ++i) {
        const long long te = (tile0 + i) * 16;
        if (te >= E) break;                       // wave-uniform exit

        // prefetch the edge_attr stream a few tiles ahead (2112B window)
        __builtin_prefetch(ea + (te + 16LL * PFD) * 33 + lane * 16, 0, 0);

        // clamped edge id: always in-bounds, so every load below is
        // unconditional (no EXEC branches -> loads cluster & overlap WMMA)
        long long e16 = te + m16;
        const long long e16s = (e16 < E) ? e16 : (E - 1);
        const int col = (int)ei[E + e16s];        // src node, always valid

        // ---- gather all 17 A-fragments into registers ----
        v2f afrag[NCHUNK];
        // x part: chunks 0..7 (K=0..31); base 8B-aligned -> b64 loads
        const v2f* __restrict__ xv = (const v2f*)(x + (long long)col * 32 + kb);
#pragma unroll
        for (int c = 0; c < 8; ++c)
            afrag[c] = xv[2 * c];
        // edge_attr part: chunks 8..15 (K=32..63); 4B-aligned scalar pairs
        const float* __restrict__ eb = ea + e16s * 33 + kb;
#pragma unroll
        for (int c = 8; c < 16; ++c) {
            afrag[c].x = eb[4 * (c - 8)];
            afrag[c].y = eb[4 * (c - 8) + 1];
        }
        // chunk 16 (K=64..67): only K=64 is real; load unconditionally,
        // kill by value-select (K=65..67 hit zero rows of bfrag anyway)
        {
            float v = ea[e16s * 33 + 32];
            afrag[16].x = (kb == 0) ? v : 0.0f;
            afrag[16].y = 0.0f;
        }

        // ---- 17 back-to-back f32 WMMAs ----
        v8f acc = {};
#pragma unroll
        for (int c = 0; c < NCHUNK; ++c)
            acc = __builtin_amdgcn_wmma_f32_16x16x4_f32(
                false, afrag[c], false, bfrag[c], (short)0, acc, false, false);

        // ---- bias + ReLU + atomic scatter ----
        // C layout: VGPR v holds edge m = v + 8*hi, column n = lane&15.
        if (n < 4) {
#pragma unroll
            for (int v = 0; v < 8; ++v) {
                long long e = te + v + 8 * hi;
                if (e < E) {
                    int   r = (int)ei[e];         // dest node (L0/L2 hit)
                    float h = fmaxf(acc[v] + bias, 0.0f);
                    atomicAdd(&out[(long long)r * 36 + 32 + n], h);
                    if (n == 0) atomicAdd(&cnt[r], 1.0f);
                }
            }
        }
    }
}

// Kernel 3: mean = sum / max(cnt, 1)
__global__ void finalize_kernel(float* __restrict__ out,
                                const float* __restrict__ cnt, int N) {
    int j = blockIdx.x * blockDim.x + threadIdx.x;
    if (j < N * 4) {
        int i = j >> 2, nn = j & 3;
        out[(long long)i * 36 + 32 + nn] /= fmaxf(cnt[i], 1.0f);
    }
}

extern "C" void kernel_launch(void* const* d_in, const int* in_sizes, int n_in,
                              void* d_out, int out_size, void* d_ws, size_t ws_size,
                              hipStream_t stream) {
    const float*     x  = (const float*)d_in[0];
    const long long* ei = (const long long*)d_in[1];   // int64 per reference
    const float*     ea = (const float*)d_in[2];
    const float*     W1 = (const float*)d_in[3];
    const float*     b1 = (const float*)d_in[4];
    float* out = (float*)d_out;

    const int       N = in_sizes[0] / 32;
    const long long E = (long long)in_sizes[2] / 33;
    float* cnt = (float*)d_ws;    // N floats of scratch (re-zeroed every launch)

    {
        int total  = N * 36;
        int blocks = (total + 255) / 256;
        init_out_kernel<<<blocks, 256, 0, stream>>>(x, out, cnt, N);
    }
    if (E > 0) {
        long long tiles  = (E + 15) / 16;
        long long blocks = (tiles + (long long)WAVES * ITER - 1) / ((long long)WAVES * ITER);
        edge_kernel<<<dim3((unsigned)blocks), dim3(WAVES * 32), 0, stream>>>(
            x, ei, ea, W1, b1, out, cnt, E);
    }
    {
        int total  = N * 4;
        int blocks = (total + 255) / 256;
        finalize_kernel<<<blocks, 256, 0, stream>>>(out, cnt, N);
    }
}